// InputRegion_31361851195853
// MI455X (gfx1250) — compile-verified
//
#include <hip/hip_runtime.h>

// ---------------------------------------------------------------------------
// Chain product of 512 (I + tiny perturbation) 128x128 f32 matrices per batch.
// f32 WMMA (v_wmma_f32_16x16x4_f32) is mandatory for precision: products are
// I + O(1e-3); f16/bf16 mantissas cannot represent that.
// Two-level tree: stage1 = 16 batches x 16 chunks (32 mats each, 31 matmuls),
// stage2 = 16 batches folding 16 chunk partials. Partials: 16 MB in d_ws.
//
// Round 3: transposed recurrence R_t = M_t^T x R_{t-1} (R = P^T), with R
// stored column-major == P row-major in the same LDS buffer. All three WMMA
// operand paths are now contiguous:
//   A-frag: ds_load_b64 from MbT (M^T row-major, built that way from global)
//   B-frag: ds_load_b64 from Pb (running product, row-major)
//   D write: 2x ds_store_b128 per tile (transposed writeback -> j contiguous)
// First K-step consumes WMMA's inline-0 SRC2 (no accumulator zero-init movs).
// ---------------------------------------------------------------------------

typedef __attribute__((ext_vector_type(2))) float v2f;
typedef __attribute__((ext_vector_type(4))) float v4f;
typedef __attribute__((ext_vector_type(8))) float v8f;

#define DM     128          // matrix dimension
#define LDP    132          // LDS row pitch in floats (132 mod 64 banks = 4)
#define SEQ    512          // chain length
#define CHUNK  32           // matrices per stage1 chunk
#define NCHUNK 16           // SEQ / CHUNK
#define NBATCH 16

// D(regs) = MbT(A = M^T) x Pb(B = R = P^T stored as P row-major).
// Block = 256 threads = 8 waves; wave w computes D tile-row w (8 16x16 tiles).
// A-frag (32-bit A 16x4): lanes 0-15 K={k,k+1}, lanes 16-31 K={k+2,k+3}.
// B-frag (4x16): vgpr0 = B[kk][n], vgpr1 = B[kk+1][n] = Pb[n*LDP + kk..kk+1].
__device__ __forceinline__ void mm128T(const float* __restrict__ MbT,
                                       const float* __restrict__ Pb,
                                       v8f acc[8]) {
  const int lane = threadIdx.x & 31;
  const int wave = threadIdx.x >> 5;       // tile row 0..7
  const int lo   = lane & 15;
  const int hi   = lane >> 4;
  const int arow = wave * 16 + lo;         // A-matrix row for this lane

  {                                         // k = 0: C = inline 0
    const int kk = 2 * hi;
    const v2f a = *(const v2f*)&MbT[arow * LDP + kk];
#pragma unroll
    for (int nj = 0; nj < 8; ++nj) {
      const v2f b = *(const v2f*)&Pb[(nj * 16 + lo) * LDP + kk];
      acc[nj] = __builtin_amdgcn_wmma_f32_16x16x4_f32(
          false, a, false, b, (short)0, (v8f)0.f, false, false);
    }
  }
#pragma unroll 4
  for (int k = 4; k < DM; k += 4) {
    const int kk = k + 2 * hi;
    const v2f a = *(const v2f*)&MbT[arow * LDP + kk];
#pragma unroll
    for (int nj = 0; nj < 8; ++nj) {
      const v2f b = *(const v2f*)&Pb[(nj * 16 + lo) * LDP + kk];
      acc[nj] = __builtin_amdgcn_wmma_f32_16x16x4_f32(
          false, a, false, b, (short)0, acc[nj], false, false);
    }
  }
}

// Transposed writeback: D = R_t, element (dr,dc) -> Pb[dc*LDP + dr].
// C/D layout: vgpr j -> tile row j (lanes 0-15) / j+8 (lanes 16-31), col lo.
// j runs over consecutive floats -> two ds_store_b128 per tile.
__device__ __forceinline__ void store_accT(float* __restrict__ Pb,
                                           const v8f acc[8]) {
  const int lane  = threadIdx.x & 31;
  const int wave  = threadIdx.x >> 5;
  const int lo    = lane & 15;
  const int hi    = lane >> 4;
  const int rbase = wave * 16 + hi * 8;
#pragma unroll
  for (int nj = 0; nj < 8; ++nj) {
    float* p = &Pb[(nj * 16 + lo) * LDP + rbase];
    const v4f l4 = __builtin_shufflevector(acc[nj], acc[nj], 0, 1, 2, 3);
    const v4f h4 = __builtin_shufflevector(acc[nj], acc[nj], 4, 5, 6, 7);
    *(v4f*)p       = l4;
    *(v4f*)(p + 4) = h4;
  }
}

// Build M_s^T row-major in LDS: MbT[r*LDP + l] = M[l][r], where
// M = i0*T[s,:,:,0] + i1*T[s,:,:,1] + I. Global float2 reads stay coalesced.
__device__ __forceinline__ void buildMT(float* __restrict__ MbT,
                                        const float2* __restrict__ t2,
                                        float i0, float i1) {
  for (int e = threadIdx.x; e < DM * DM; e += 256) {
    const int l = e >> 7, r = e & 127;
    const float2 t = t2[e];                    // (i=0, i=1) adjacent in memory
    MbT[r * LDP + l] = fmaf(i0, t.x, i1 * t.y) + ((l == r) ? 1.f : 0.f);
  }
}

// Load a row-major global matrix into LDS transposed (for the A side).
__device__ __forceinline__ void loadMT(float* __restrict__ MbT,
                                       const float* __restrict__ g) {
  for (int e = threadIdx.x; e < DM * DM; e += 256)
    MbT[(e & 127) * LDP + (e >> 7)] = g[e];
}

// Load a row-major global matrix into LDS row-major (initial running product).
__device__ __forceinline__ void loadM(float* __restrict__ Mb,
                                      const float* __restrict__ g) {
  for (int e = threadIdx.x; e < DM * DM; e += 256)
    Mb[(e >> 7) * LDP + (e & 127)] = g[e];
}

// ---- Stage 1: one workgroup per (batch, chunk). Product of 32 matrices.
__global__ __launch_bounds__(256)
void chain_stage1(const float* __restrict__ input,   // (B, S, 2)
                  const float* __restrict__ tensor,  // (S, 128, 128, 2)
                  float* __restrict__ partials) {    // (B, NCHUNK, 128, 128)
  __shared__ float Pb[DM * LDP];    // running product P, row-major
  __shared__ float MbT[DM * LDP];   // next matrix, transposed (A side)
  const int c  = blockIdx.x;
  const int b  = blockIdx.y;
  const int s0 = c * CHUNK;

  // P = M_{s0}, row-major.
  {
    const float i0 = input[(b * SEQ + s0) * 2 + 0];
    const float i1 = input[(b * SEQ + s0) * 2 + 1];
    const float2* t2 = (const float2*)tensor + (size_t)s0 * DM * DM;
    for (int e = threadIdx.x; e < DM * DM; e += 256) {
      const int l = e >> 7, r = e & 127;
      const float2 t = t2[e];
      Pb[l * LDP + r] = fmaf(i0, t.x, i1 * t.y) + ((l == r) ? 1.f : 0.f);
    }
  }
  __syncthreads();

  v8f acc[8];
  for (int t = 1; t < CHUNK; ++t) {
    const int s = s0 + t;
    buildMT(MbT, (const float2*)tensor + (size_t)s * DM * DM,
            input[(b * SEQ + s) * 2 + 0], input[(b * SEQ + s) * 2 + 1]);
    if (t + 1 < CHUNK) {           // prefetch next matrix (global_prefetch_b8)
      const char* nxt = (const char*)tensor + (size_t)(s + 1) * DM * DM * 8;
#pragma unroll
      for (int q = 0; q < 8; ++q)
        __builtin_prefetch(nxt + ((size_t)threadIdx.x + q * 256) * 64, 0, 3);
    }
    __syncthreads();
    mm128T(MbT, Pb, acc);          // R_t = M_s^T x R_{t-1}  (== P @ M_s)
    __syncthreads();
    store_accT(Pb, acc);           // P <- P @ M_s (row-major via transpose)
    __syncthreads();
  }

  float* out = partials + ((size_t)b * NCHUNK + c) * (DM * DM);
  for (int e = threadIdx.x; e < DM * DM; e += 256)
    out[e] = Pb[(e >> 7) * LDP + (e & 127)];
}

// ---- Stage 2: one workgroup per batch. Fold 16 chunk products in order.
__global__ __launch_bounds__(256)
void chain_stage2(const float* __restrict__ partials,
                  float* __restrict__ out) {
  __shared__ float Pb[DM * LDP];    // running product, row-major
  __shared__ float MbT[DM * LDP];   // chunk matrix, transposed (A side)
  const int b = blockIdx.x;
  const float* base = partials + (size_t)b * NCHUNK * (DM * DM);

  loadM(Pb, base);
  __syncthreads();

  v8f acc[8];
  for (int c = 1; c < NCHUNK; ++c) {
    loadMT(MbT, base + (size_t)c * (DM * DM));
    __syncthreads();
    mm128T(MbT, Pb, acc);
    __syncthreads();
    store_accT(Pb, acc);
    __syncthreads();
  }

  float* o = out + (size_t)b * (DM * DM);
  for (int e = threadIdx.x; e < DM * DM; e += 256)
    o[e] = Pb[(e >> 7) * LDP + (e & 127)];
}

extern "C" void kernel_launch(void* const* d_in, const int* in_sizes, int n_in,
                              void* d_out, int out_size, void* d_ws, size_t ws_size,
                              hipStream_t stream) {
  const float* input  = (const float*)d_in[0];   // (16, 512, 2) f32
  const float* tensor = (const float*)d_in[1];   // (512, 128, 128, 2) f32
  // d_in[2] = bias_mat (identity) -- fused into matrix construction.
  float* partials = (float*)d_ws;                // 16*16*128*128*4 = 16 MB
  float* outp     = (float*)d_out;               // 16*128*128 f32

  dim3 g1(NCHUNK, NBATCH);
  chain_stage1<<<g1, 256, 0, stream>>>(input, tensor, partials);
  chain_stage2<<<NBATCH, 256, 0, stream>>>(partials, outp);
}